// RNA_embedding_11304353923370
// MI455X (gfx1250) — compile-verified
//
#include <hip/hip_runtime.h>

typedef float v2f __attribute__((ext_vector_type(2)));
typedef float v4f __attribute__((ext_vector_type(4)));
typedef float v8f __attribute__((ext_vector_type(8)));

// ---------------------------------------------------------------------------
// Kernel 1: q[which][v][e] = sum_d emb[v][d] * W[e][which*128 + d]  (+ bias
// folded into which==0).  V=5 rows padded to M=16.  One block, 16 waves:
// wave = which*8 + ntile ; each wave owns a 16x16 output tile and runs a
// K-loop of 32 V_WMMA_F32_16X16X4_F32 steps.
// ---------------------------------------------------------------------------
__global__ __launch_bounds__(512) void qgemm_kernel(const float* __restrict__ emb,
                                                    const float* __restrict__ W,
                                                    const float* __restrict__ bias,
                                                    float* __restrict__ q) {
    const int tid   = threadIdx.x;
    const int lane  = tid & 31;
    const int wave  = tid >> 5;        // 0..15
    const int which = wave >> 3;       // 0 -> W1, 1 -> W2
    const int ntile = wave & 7;        // e-tile index
    const int half  = lane >> 4;       // lane group (0: lanes 0-15, 1: 16-31)
    const int n     = lane & 15;
    const int e     = ntile * 16 + n;  // output column (0..127)
    const int m     = lane & 15;       // A-matrix row for this lane
    const int mc    = (m < 5) ? m : 4; // clamped row (avoid OOB; masked below)

    v8f acc = {0.f, 0.f, 0.f, 0.f, 0.f, 0.f, 0.f, 0.f};

#if __has_builtin(__builtin_amdgcn_wmma_f32_16x16x4_f32)
    const float* wrow = W + (size_t)e * 256 + which * 128;
    for (int kk = 0; kk < 128; kk += 4) {
        const int kb = kk + 2 * half;           // ISA layout: VGPR0->K=kb, VGPR1->K=kb+1
        float ax = emb[mc * 128 + kb];
        float ay = emb[mc * 128 + kb + 1];
        v2f a, bf;
        a.x = (m < 5) ? ax : 0.0f;              // cndmask, keeps EXEC all-1s for WMMA
        a.y = (m < 5) ? ay : 0.0f;
        bf.x = wrow[kb];
        bf.y = wrow[kb + 1];
        acc = __builtin_amdgcn_wmma_f32_16x16x4_f32(
            /*neg_a=*/false, a, /*neg_b=*/false, bf,
            /*c_mod=*/(short)0, acc, /*reuse_a=*/false, /*reuse_b=*/false);
    }
#else
    // Scalar fallback producing the same C/D VGPR layout.
    #pragma unroll
    for (int r = 0; r < 8; ++r) {
        int mr = r + half * 8;
        if (mr < 5) {
            float sum = 0.f;
            for (int d = 0; d < 128; ++d)
                sum += emb[mr * 128 + d] * W[(size_t)e * 256 + which * 128 + d];
            acc[r] = sum;
        }
    }
#endif

    // C/D layout: VGPR r -> M = r + 8*half, N = lane&15.  Only rows M<5 are real.
    const float bb = bias[e];
    #pragma unroll
    for (int r = 0; r < 8; ++r) {
        int mr = r + half * 8;
        if (mr < 5) {
            float val = acc[r] + (which == 0 ? bb : 0.0f);
            q[which * 640 + mr * 128 + e] = val;   // q: [2][5][128]
        }
    }
}

// ---------------------------------------------------------------------------
// Kernel 2: s[row][:] = emb[x[row]][:]   (row = b*1024 + l), float4 copies.
// 256 blocks x 256 threads; 8 rows per block, 32 lanes (float4 each) per row.
// ---------------------------------------------------------------------------
__global__ __launch_bounds__(256) void embed_kernel(const int* __restrict__ x,
                                                    const float* __restrict__ emb,
                                                    float* __restrict__ s) {
    const int tid = threadIdx.x;
    const int row = blockIdx.x * 8 + (tid >> 5);   // 0..2047
    const int t   = tid & 31;
    const int v   = x[row];
    const v4f* src = (const v4f*)(emb + (size_t)v * 128);
    v4f*       dst = (v4f*)(s + (size_t)row * 128);
    dst[t] = src[t];
}

// ---------------------------------------------------------------------------
// Kernel 3: m[b][e][i][j] = q1[x[b,i]][e] + q2[x[b,j]][e]
// One block per (b, e, 8-row i-chunk).  Thread t loads x[b, 4t..4t+3] once,
// gathers its 4 q2 values once (L1 hits), then streams 8 float4 NT stores.
// grid = 2*128*128 = 32768 blocks of 256 threads; 32 KB written per block.
// ---------------------------------------------------------------------------
__global__ __launch_bounds__(256) void bcast_kernel(const int* __restrict__ x,
                                                    const float* __restrict__ q,
                                                    float* __restrict__ mout) {
    const int gid = blockIdx.x;
    const int ic  = gid & 127;          // i-chunk (0..127), 8 rows each
    const int be  = gid >> 7;
    const int e   = be & 127;
    const int b   = be >> 7;
    const int i0  = ic * 8;
    const int t   = threadIdx.x;        // 0..255 -> j = 4t..4t+3

    const float* q1 = q;                // bias already folded in
    const float* q2 = q + 640;
    const int*   xb = x + b * 1024;

    const int4 xj = ((const int4*)xb)[t];
    const float w0 = q2[xj.x * 128 + e];
    const float w1 = q2[xj.y * 128 + e];
    const float w2 = q2[xj.z * 128 + e];
    const float w3 = q2[xj.w * 128 + e];

    const size_t base = (((size_t)b * 128 + e) * 1024 + i0) * 1024;
    #pragma unroll
    for (int r = 0; r < 8; ++r) {
        const float vi = q1[xb[i0 + r] * 128 + e];
        v4f o = { vi + w0, vi + w1, vi + w2, vi + w3 };
        __builtin_nontemporal_store(o, (v4f*)(mout + base + (size_t)r * 1024 + t * 4));
    }
}

extern "C" void kernel_launch(void* const* d_in, const int* in_sizes, int n_in,
                              void* d_out, int out_size, void* d_ws, size_t ws_size,
                              hipStream_t stream) {
    const int*   x    = (const int*)d_in[0];     // (2,1024) int32
    const float* emb  = (const float*)d_in[1];   // (5,128)
    const float* W    = (const float*)d_in[2];   // (128,256)
    const float* bias = (const float*)d_in[3];   // (128,)

    float* out = (float*)d_out;
    float* s   = out;                 // 2*1024*128 = 262144 floats
    float* m   = out + 262144;        // 2*128*1024*1024 floats
    float* q   = (float*)d_ws;        // 2*5*128 = 1280 floats of scratch

    qgemm_kernel<<<1, 512, 0, stream>>>(emb, W, bias, q);
    embed_kernel<<<256, 256, 0, stream>>>(x, emb, s);
    bcast_kernel<<<32768, 256, 0, stream>>>(x, q, m);
}